// TTEmbeddingLayer_84713934946739
// MI455X (gfx1250) — compile-verified
//
#include <hip/hip_runtime.h>

// ---------------------------------------------------------------------------
// TT-embedding for MI455X (gfx1250, wave32).
//
// out[t, e] = sum_k left[r_t, k] * core2[k*4e6 + cbase_t + e],  r_t = v/62500
// cbase_t = v*64 - r_t*4e6   (valid because 4e6 % 64 == 0: a token's 64
// elements never cross a row boundary of the 8 x 4e6 matrix).
//
// Traffic: ~16.8 MB gathered reads + 2 MB writes  => ~0.8 us @ 23.3 TB/s.
// FLOPs: 8.4 MFLOP -> totally memory bound; never materialize the 128 MB table.
// ---------------------------------------------------------------------------

typedef __attribute__((ext_vector_type(2))) float v2f;
typedef __attribute__((ext_vector_type(8))) float v8f;

// Prep kernel (1 wave): left2(2x32) = core0(2x8, zero-padded to 16x8)
//                                   @ core1.reshape(8x32)
// via V_WMMA_F32_16X16X4_F32, accumulating two K=4 steps, two N=16 tiles.
// left2[i, j*8+c] == left[(i*4+j), c] from the reference einsum.
__global__ void __launch_bounds__(32) tt_left_wmma(
    const float* __restrict__ core0,   // 16 floats  (1,2,8)
    const float* __restrict__ core1,   // 256 floats (8,4,8) == (8,32) row-major
    float* __restrict__ left2)         // 64 floats out (2 rows x 32)
{
  const int lane = threadIdx.x;        // 0..31, EXEC all-1s for WMMA
  const int m    = lane & 15;          // A-row / B-col within half-wave
  const int khi  = (lane >> 4) << 1;   // lanes 0-15 -> K{0,1}; 16-31 -> K{2,3}

  #pragma unroll
  for (int tile = 0; tile < 2; ++tile) {
    const int n0 = tile * 16;
    v8f acc = {0.f, 0.f, 0.f, 0.f, 0.f, 0.f, 0.f, 0.f};
    #pragma unroll
    for (int s = 0; s < 2; ++s) {      // K = 8 as two K=4 WMMA steps
      const int k0 = s * 4 + khi;
      v2f a, b;
      // A (16x4): rows 0,1 = core0, rows 2..15 zero padding.
      a.x = (m < 2) ? core0[m * 8 + k0]     : 0.0f;
      a.y = (m < 2) ? core0[m * 8 + k0 + 1] : 0.0f;
      // B (4x16): B[k, n] = core1[k*32 + n0 + n]
      b.x = core1[(k0    ) * 32 + n0 + m];
      b.y = core1[(k0 + 1) * 32 + n0 + m];
      acc = __builtin_amdgcn_wmma_f32_16x16x4_f32(
          /*neg_a=*/false, a, /*neg_b=*/false, b,
          /*c_mod=*/(short)0, acc, /*reuse_a=*/false, /*reuse_b=*/false);
    }
    // D layout (32-bit C/D 16x16): lanes 0-15 hold N=lane; VGPR v = row M=v.
    if (lane < 16) {
      left2[ 0 + n0 + lane] = acc[0];  // row i=0
      left2[32 + n0 + lane] = acc[1];  // row i=1
    }
  }
}

// Main bandwidth-bound gather-contract. One thread = one float4 of output.
// 16 consecutive threads cover one token -> each core2 row access is a
// contiguous 256B burst of global_load_b128.
__global__ void __launch_bounds__(256) tt_gather(
    const int* __restrict__ ids,       // 8192 token ids
    const float* __restrict__ core2,   // 8 x 4,000,000 floats
    const float* __restrict__ left2,   // 64 floats (from prep kernel)
    float* __restrict__ out,           // 8192 x 64 floats
    int total)                         // tokens * 16
{
  __shared__ float sl[64];
  if (threadIdx.x < 64) sl[threadIdx.x] = left2[threadIdx.x];
  __syncthreads();

  const int gid = blockIdx.x * 256 + threadIdx.x;
  if (gid >= total) return;

  const int t = gid >> 4;              // token
  const int q = gid & 15;              // float4 index within the 64-wide row
  const int v = ids[t];
  const int r = (int)((unsigned)v / 62500u);          // row of 8x4M matrix
  const int cbase = v * 64 - r * 4000000 + q * 4;     // (v*64) % 4e6 + e

  const float4* __restrict__ b0 = (const float4*)(core2 + cbase);
  // left[r, k] == left2[(r>>2)*32 + (r&3)*8 + k]
  const float* __restrict__ lr = sl + ((r >> 2) * 32 + (r & 3) * 8);

  float4 acc; acc.x = acc.y = acc.z = acc.w = 0.0f;
  #pragma unroll
  for (int k = 0; k < 8; ++k) {
    const float4 bb = b0[k * 1000000];  // stride 4e6 floats = 1e6 float4s
    const float  w  = lr[k];
    acc.x = fmaf(w, bb.x, acc.x);
    acc.y = fmaf(w, bb.y, acc.y);
    acc.z = fmaf(w, bb.z, acc.z);
    acc.w = fmaf(w, bb.w, acc.w);
  }
  ((float4*)out)[gid] = acc;
}

extern "C" void kernel_launch(void* const* d_in, const int* in_sizes, int n_in,
                              void* d_out, int out_size, void* d_ws, size_t ws_size,
                              hipStream_t stream) {
  const int*   ids   = (const int*)d_in[0];    // input_ids (8192)
  const float* core0 = (const float*)d_in[1];  // 16
  const float* core1 = (const float*)d_in[2];  // 256
  const float* core2 = (const float*)d_in[3];  // 32,000,000
  float* out   = (float*)d_out;
  float* left2 = (float*)d_ws;                 // 64 floats of scratch

  const int tokens = in_sizes[0];
  const int total  = tokens * 16;              // one thread per float4

  tt_left_wmma<<<1, 32, 0, stream>>>(core0, core1, left2);
  tt_gather<<<(total + 255) / 256, 256, 0, stream>>>(ids, core2, left2, out, total);
}